// OptimalTransportSimilarity_62002147885445
// MI455X (gfx1250) — compile-verified
//
#include <hip/hip_runtime.h>
#include <hip/hip_bf16.h>
#include <cstddef>

// ---------------------------------------------------------------------------
// Optimal-transport (Sinkhorn) similarity for MI455X / gfx1250.
// B=16, N=M=1024, D=64. K matrix (64MB f32) is L2-resident (192MB L2), so
// the 100 Sinkhorn iterations are L2-bandwidth bound: all streaming loops use
// float4 (global_load_b128) so each wave moves 512B per VMEM instruction.
// The one-time cost GEMM uses V_WMMA_F32_16X16X4_F32 (f32 = reference
// numerics), 16 WMMA per 16x16 tile over D=64.
// ---------------------------------------------------------------------------

#define B_      16
#define N_      1024
#define M_      1024
#define D_      64
#define INV_EPS_ 10.0f          // 1/EPSILON, EPSILON = 0.1
#define EPS_DIV_ 1e-8f
#define MAX_ITER_ 100
#define NSPLIT_  16             // n-range splits for the v-update partials
#define A_VAL_  (1.0f / (float)N_)   // a = (1/N)/(1+1e-8) == 1/N in f32
#define B_VAL_  (1.0f / (float)M_)

typedef float v2f __attribute__((ext_vector_type(2)));
typedef float v8f __attribute__((ext_vector_type(8)));

// ---- init: row squared-norms + u0/v0 --------------------------------------
__global__ void ot_init_kernel(const float* __restrict__ f1, const float* __restrict__ f2,
                               float* __restrict__ xx, float* __restrict__ yy,
                               float* __restrict__ u, float* __restrict__ v) {
    int idx = blockIdx.x * blockDim.x + threadIdx.x;   // 32768 threads total
    if (idx < B_ * N_) {
        const float4* p = (const float4*)(f1 + (size_t)idx * D_);
        float s = 0.f;
#pragma unroll
        for (int d = 0; d < D_ / 4; ++d) {
            float4 q = p[d];
            s += q.x * q.x + q.y * q.y + q.z * q.z + q.w * q.w;
        }
        xx[idx] = s;
        u[idx]  = 1.0f / (float)N_;
    } else {
        int j = idx - B_ * N_;
        const float4* p = (const float4*)(f2 + (size_t)j * D_);
        float s = 0.f;
#pragma unroll
        for (int d = 0; d < D_ / 4; ++d) {
            float4 q = p[d];
            s += q.x * q.x + q.y * q.y + q.z * q.z + q.w * q.w;
        }
        yy[j] = s;
        v[j]  = 1.0f / (float)M_;
    }
}

// ---- cost + K: one 16x16 output tile per wave via V_WMMA_F32_16X16X4_F32 ---
// A-frag (16x4 f32, ISA 7.12.2): lane l -> M = l&15, K-pair = 2*(l>>4);
// B-frag (4x16) mirrors it from features2 (K striped across VGPRs).
__global__ void ot_cost_kernel(const float* __restrict__ f1, const float* __restrict__ f2,
                               const float* __restrict__ xx, const float* __restrict__ yy,
                               float* __restrict__ C, float* __restrict__ K) {
    int wave = blockIdx.x * (blockDim.x >> 5) + (threadIdx.x >> 5);
    int lane = threadIdx.x & 31;
    const int TPB = (N_ / 16) * (M_ / 16);             // 4096 tiles per batch
    int b  = wave / TPB;
    int t  = wave % TPB;
    int n0 = (t / (M_ / 16)) * 16;
    int m0 = (t % (M_ / 16)) * 16;

    int half = lane >> 4;
    int lr   = lane & 15;

    const float* arow = f1 + ((size_t)(b * N_ + n0 + lr)) * D_ + 2 * half;
    const float* brow = f2 + ((size_t)(b * M_ + m0 + lr)) * D_ + 2 * half;

    v8f acc = {};
#pragma unroll
    for (int k = 0; k < D_; k += 4) {
        v2f a  = *(const v2f*)(arow + k);
        v2f bb = *(const v2f*)(brow + k);
        // D = A(16x4) * B(4x16) + C ; 8 args: neg_a, A, neg_b, B, c_mod, C, reuse_a, reuse_b
        acc = __builtin_amdgcn_wmma_f32_16x16x4_f32(false, a, false, bb,
                                                    (short)0, acc, false, false);
    }

    float yv = yy[b * M_ + m0 + lr];
#pragma unroll
    for (int r = 0; r < 8; ++r) {
        int   row   = n0 + r + 8 * half;              // C/D layout: VGPR r -> M=r (lanes<16), r+8
        float cost2 = xx[b * N_ + row] + yv - 2.0f * acc[r];
        float cv    = sqrtf(fmaxf(cost2, 1e-10f));
        size_t off  = ((size_t)(b * N_ + row)) * M_ + (m0 + lr);
        C[off] = cv;
        K[off] = __expf(-cv * INV_EPS_);              // underflows to 0 like f32 reference
    }
}

// ---- u-update: one wave per row, float4 streams, shuffle-reduced -----------
__global__ void ot_u_kernel(const float* __restrict__ K, const float* __restrict__ v,
                            float* __restrict__ u) {
    int wave = blockIdx.x * (blockDim.x >> 5) + (threadIdx.x >> 5);
    int lane = threadIdx.x & 31;
    int b = wave / N_, n = wave % N_;
    const float4* Kr = (const float4*)(K + ((size_t)(b * N_ + n)) * M_);
    const float4* vb = (const float4*)(v + (size_t)b * M_);
    float4 a4 = make_float4(0.f, 0.f, 0.f, 0.f);
#pragma unroll
    for (int i = lane; i < M_ / 4; i += 32) {          // 8 x b128 per lane
        float4 k4 = Kr[i];
        float4 v4 = vb[i];
        a4.x += k4.x * v4.x; a4.y += k4.y * v4.y;
        a4.z += k4.z * v4.z; a4.w += k4.w * v4.w;
    }
    float acc = (a4.x + a4.y) + (a4.z + a4.w);
    for (int off = 16; off > 0; off >>= 1) acc += __shfl_xor(acc, off, 32);
    if (lane == 0) u[b * N_ + n] = A_VAL_ / (acc + EPS_DIV_);
}

// ---- v-update stage 1: each thread owns 4 columns (b128), no atomics -------
__global__ void ot_vpart_kernel(const float* __restrict__ K, const float* __restrict__ u,
                                float* __restrict__ partial) {
    int tid = threadIdx.x;                              // 256 threads = 1024 columns
    int s   = blockIdx.x % NSPLIT_;
    int b   = blockIdx.x / NSPLIT_;
    int n0  = s * (N_ / NSPLIT_);                       // 64 rows per split
    const float4* Kp = (const float4*)(K + ((size_t)(b * N_ + n0)) * M_) + tid;
    const float*  ub = u + b * N_ + n0;
    float4 acc = make_float4(0.f, 0.f, 0.f, 0.f);
#pragma unroll 4
    for (int i = 0; i < N_ / NSPLIT_; ++i) {
        float4 k4 = Kp[(size_t)i * (M_ / 4)];
        float  uu = ub[i];                              // uniform -> scalarized
        acc.x += k4.x * uu; acc.y += k4.y * uu;
        acc.z += k4.z * uu; acc.w += k4.w * uu;
    }
    ((float4*)partial)[(size_t)(b * NSPLIT_ + s) * (M_ / 4) + tid] = acc;
}

// ---- v-update stage 2: fixed-order partial reduction (deterministic) -------
__global__ void ot_vfin_kernel(const float* __restrict__ partial, float* __restrict__ v) {
    int idx = blockIdx.x * blockDim.x + threadIdx.x;   // B*M/4 threads
    int b  = idx / (M_ / 4);
    int m4 = idx % (M_ / 4);
    float4 s = make_float4(0.f, 0.f, 0.f, 0.f);
#pragma unroll
    for (int i = 0; i < NSPLIT_; ++i) {
        float4 p = ((const float4*)partial)[(size_t)(b * NSPLIT_ + i) * (M_ / 4) + m4];
        s.x += p.x; s.y += p.y; s.z += p.z; s.w += p.w;
    }
    float4 r;
    r.x = B_VAL_ / (s.x + EPS_DIV_);
    r.y = B_VAL_ / (s.y + EPS_DIV_);
    r.z = B_VAL_ / (s.z + EPS_DIV_);
    r.w = B_VAL_ / (s.w + EPS_DIV_);
    ((float4*)v)[idx] = r;
}

// ---- final: row_loss[b,n] = u * sum_m K*C*v --------------------------------
__global__ void ot_loss_kernel(const float* __restrict__ K, const float* __restrict__ C,
                               const float* __restrict__ u, const float* __restrict__ v,
                               float* __restrict__ row_loss) {
    int wave = blockIdx.x * (blockDim.x >> 5) + (threadIdx.x >> 5);
    int lane = threadIdx.x & 31;
    int b = wave / N_, n = wave % N_;
    const float4* Kr = (const float4*)(K + ((size_t)(b * N_ + n)) * M_);
    const float4* Cr = (const float4*)(C + ((size_t)(b * N_ + n)) * M_);
    const float4* vb = (const float4*)(v + (size_t)b * M_);
    float4 a4 = make_float4(0.f, 0.f, 0.f, 0.f);
#pragma unroll
    for (int i = lane; i < M_ / 4; i += 32) {
        float4 k4 = Kr[i];
        float4 c4 = Cr[i];
        float4 v4 = vb[i];
        a4.x += k4.x * c4.x * v4.x; a4.y += k4.y * c4.y * v4.y;
        a4.z += k4.z * c4.z * v4.z; a4.w += k4.w * c4.w * v4.w;
    }
    float acc = (a4.x + a4.y) + (a4.z + a4.w);
    for (int off = 16; off > 0; off >>= 1) acc += __shfl_xor(acc, off, 32);
    if (lane == 0) row_loss[b * N_ + n] = u[b * N_ + n] * acc;
}

// ---- single-block deterministic reduction -> mean over batch ---------------
__global__ void ot_reduce_kernel(const float* __restrict__ row_loss, float* __restrict__ out) {
    __shared__ float sm[1024];
    int tid = threadIdx.x;
    float s = 0.f;
    for (int i = tid; i < B_ * N_; i += 1024) s += row_loss[i];
    sm[tid] = s;
    __syncthreads();
    for (int stride = 512; stride > 0; stride >>= 1) {
        if (tid < stride) sm[tid] += sm[tid + stride];
        __syncthreads();
    }
    if (tid == 0) out[0] = sm[0] / (float)B_;
}

extern "C" void kernel_launch(void* const* d_in, const int* in_sizes, int n_in,
                              void* d_out, int out_size, void* d_ws, size_t ws_size,
                              hipStream_t stream) {
    const float* f1 = (const float*)d_in[0];   // [B,N,D]
    const float* f2 = (const float*)d_in[1];   // [B,M,D]
    float* out = (float*)d_out;

    // workspace carve-up (~130 MB of floats); all offsets float4-aligned
    float* ws = (float*)d_ws;
    size_t o = 0;
    float* K        = ws + o; o += (size_t)B_ * N_ * M_;        // 64 MB
    float* C        = ws + o; o += (size_t)B_ * N_ * M_;        // 64 MB
    float* xx       = ws + o; o += (size_t)B_ * N_;
    float* yy       = ws + o; o += (size_t)B_ * M_;
    float* u        = ws + o; o += (size_t)B_ * N_;
    float* v        = ws + o; o += (size_t)B_ * M_;
    float* partial  = ws + o; o += (size_t)B_ * NSPLIT_ * M_;   // 1 MB
    float* row_loss = ws + o; o += (size_t)B_ * N_;
    (void)ws_size; (void)in_sizes; (void)n_in; (void)out_size;

    ot_init_kernel<<<(B_ * (N_ + M_)) / 256, 256, 0, stream>>>(f1, f2, xx, yy, u, v);

    // 65536 tiles, 8 waves (one tile each) per 256-thread block
    ot_cost_kernel<<<(B_ * (N_ / 16) * (M_ / 16)) / 8, 256, 0, stream>>>(f1, f2, xx, yy, C, K);

    for (int it = 0; it < MAX_ITER_; ++it) {
        ot_u_kernel<<<(B_ * N_) / 8, 256, 0, stream>>>(K, v, u);
        ot_vpart_kernel<<<B_ * NSPLIT_, 256, 0, stream>>>(K, u, partial);
        ot_vfin_kernel<<<(B_ * M_ / 4) / 256, 256, 0, stream>>>(partial, v);
    }

    ot_loss_kernel<<<(B_ * N_) / 8, 256, 0, stream>>>(K, C, u, v, row_loss);
    ot_reduce_kernel<<<1, 1024, 0, stream>>>(row_loss, out);
}